// GNN_58274116272537
// MI455X (gfx1250) — compile-verified
//
#include <hip/hip_runtime.h>
#include <hip/hip_bf16.h>

typedef __attribute__((ext_vector_type(2))) float v2f;
typedef __attribute__((ext_vector_type(8))) float v8f;

#define IN_CH   64
#define HID_CH  128
#define BN_EPS  1e-5f
#define WPAD    132   // LDS column stride (floats): 132 % 64 == 4 -> conflict-free b64 reads

// ---------------------------------------------------------------------------
// 0) zero scratch that is accumulated into (agg1 + stats)
// ---------------------------------------------------------------------------
__global__ void zero_ws_kernel(float* __restrict__ agg, float* __restrict__ stats, int nAgg) {
    int i = blockIdx.x * blockDim.x + threadIdx.x;
    if (i < nAgg) agg[i] = 0.0f;
    if (i < 512)  stats[i] = 0.0f;
}

// ---------------------------------------------------------------------------
// 1) agg1[dst, c] += x[src, c]   (64 threads per edge; L2-resident operands)
//    e is wave-uniform -> scalarize the edge-index fetch (s_load) so the
//    vector path only carries the gather + atomic.
// ---------------------------------------------------------------------------
__global__ void scatter_feat_kernel(const float* __restrict__ x, const int* __restrict__ ei,
                                    float* __restrict__ agg, long long E) {
    unsigned wbase = __builtin_amdgcn_readfirstlane(threadIdx.x) & ~31u;  // wave-uniform
    long long idx_w = (long long)blockIdx.x * blockDim.x + wbase;
    long long e = idx_w >> 6;                        // uniform across the wave
    if (e >= E) return;
    int c = (int)(idx_w & 63) + (int)(threadIdx.x & 31);
    int src = ei[e];                                 // scalar loads (uniform address)
    int dst = ei[E + e];
    atomicAdd(&agg[(size_t)dst * IN_CH + c], x[(size_t)src * IN_CH + c]);
}

// ---------------------------------------------------------------------------
// 2) hpre = [agg1 | x] @ [w_rel1 ; w_root1] + b_rel1   via V_WMMA_F32_16X16X4_F32
//    block = 256 threads (8 waves). Each wave owns a 16-row stripe and all
//    8 column tiles (N=128). W staged TRANSPOSED in LDS: sWt[col*WPAD + k],
//    so each B fragment is one contiguous ds_load_b64 (immediate offset).
// ---------------------------------------------------------------------------
__global__ void gemm1_wmma_kernel(const float* __restrict__ agg, const float* __restrict__ x,
                                  const float* __restrict__ w_rel1, const float* __restrict__ w_root1,
                                  const float* __restrict__ b_rel1, float* __restrict__ hpre, int n) {
    __shared__ float sWt[128 * WPAD];   // [col][k], col-padded

    int tid = threadIdx.x;
    for (int idx = tid; idx < 128 * 128; idx += 256) {
        int k   = idx >> 7;       // input channel
        int col = idx & 127;      // output channel
        float v = (k < IN_CH) ? w_rel1[idx] : w_root1[idx - IN_CH * HID_CH];
        sWt[col * WPAD + k] = v;
    }
    __syncthreads();

    const int wave = tid >> 5;
    const int lane = tid & 31;
    const int rowTile = blockIdx.x * 128 + wave * 16;  // 16 rows for this wave

    // A-fragment (32-bit A 16x4): lanes 0-15 -> K=kb,kb+1 ; lanes 16-31 -> K=kb+2,kb+3
    const int m  = lane & 15;
    const int kg = (lane >> 4) * 2;
    int rowA = rowTile + m;
    if (rowA >= n) rowA = n - 1;                       // clamp (stores are masked)
    const float* aBase = agg + (size_t)rowA * IN_CH + kg;
    const float* xBase = x   + (size_t)rowA * IN_CH + kg;

    const int ncol = lane & 15;
    const float* bBase = sWt + ncol * WPAD + kg;       // + t*16*WPAD + kk*4 are immediates

    v8f acc[8];
#pragma unroll
    for (int t = 0; t < 8; ++t) acc[t] = (v8f){0, 0, 0, 0, 0, 0, 0, 0};

    // --- K = 0..63 : neighbor aggregate through w_rel1 rows ---
#pragma unroll 4
    for (int kk = 0; kk < 16; ++kk) {
        v2f a = *(const v2f*)(aBase + kk * 4);
#pragma unroll
        for (int t = 0; t < 8; ++t) {
            v2f b = *(const v2f*)(bBase + t * 16 * WPAD + kk * 4);
            acc[t] = __builtin_amdgcn_wmma_f32_16x16x4_f32(
                false, a, false, b, (short)0, acc[t], false, false);
        }
    }
    // --- K = 64..127 : self features through w_root1 rows ---
#pragma unroll 4
    for (int kk = 0; kk < 16; ++kk) {
        v2f a = *(const v2f*)(xBase + kk * 4);
#pragma unroll
        for (int t = 0; t < 8; ++t) {
            v2f b = *(const v2f*)(bBase + t * 16 * WPAD + (16 + kk) * 4);
            acc[t] = __builtin_amdgcn_wmma_f32_16x16x4_f32(
                false, a, false, b, (short)0, acc[t], false, false);
        }
    }

    // C/D layout: VGPR r -> row (r + 8*(lane>>4)), col = lane&15 (per 16-tile)
    const int mhi = (lane >> 4) * 8;
#pragma unroll
    for (int t = 0; t < 8; ++t) {
        const int col = t * 16 + ncol;
        const float bias = b_rel1[col];
#pragma unroll
        for (int r = 0; r < 8; ++r) {
            int rr = rowTile + mhi + r;
            if (rr < n) hpre[(size_t)rr * HID_CH + col] = acc[t][r] + bias;
        }
    }
}

// ---------------------------------------------------------------------------
// 3) per-channel sum / sum-of-squares over hpre (grid-stride, LDS partials)
// ---------------------------------------------------------------------------
__global__ void bn_stats_kernel(const float* __restrict__ hpre, float* __restrict__ stats, int n) {
    __shared__ float ls[256], ls2[256];
    int tid = threadIdx.x;
    int c = tid & 127;
    int half = tid >> 7;
    float s = 0.0f, s2 = 0.0f;
    for (int row = blockIdx.x * 2 + half; row < n; row += gridDim.x * 2) {
        float v = hpre[(size_t)row * HID_CH + c];
        s += v;
        s2 += v * v;
    }
    ls[tid] = s;
    ls2[tid] = s2;
    __syncthreads();
    if (tid < 128) {
        atomicAdd(&stats[c],       ls[tid]  + ls[tid + 128]);
        atomicAdd(&stats[128 + c], ls2[tid] + ls2[tid + 128]);
    }
}

// ---------------------------------------------------------------------------
// 4) fold BN into per-channel scale/shift
// ---------------------------------------------------------------------------
__global__ void bn_finalize_kernel(float* __restrict__ stats, const float* __restrict__ gamma,
                                   const float* __restrict__ beta, int n) {
    int c = threadIdx.x;  // 128 threads
    float inv_n = 1.0f / (float)n;
    float mean = stats[c] * inv_n;
    float var  = stats[128 + c] * inv_n - mean * mean;
    float sc   = gamma[c] * rsqrtf(var + BN_EPS);
    stats[256 + c] = sc;                  // scale
    stats[384 + c] = beta[c] - mean * sc; // shift
}

// ---------------------------------------------------------------------------
// 5) y = relu(scale*hpre+shift); s_rel = y.w_rel2 ; out = y.w_root2 + b_rel2
//    one wave per node, 4 channels per lane, shuffle reduction
// ---------------------------------------------------------------------------
__global__ void bn_relu_proj_kernel(const float* __restrict__ hpre, const float* __restrict__ stats,
                                    const float* __restrict__ w_rel2, const float* __restrict__ w_root2,
                                    const float* __restrict__ b_rel2, float* __restrict__ s_rel,
                                    float* __restrict__ out, int n) {
    int wave = threadIdx.x >> 5;
    int lane = threadIdx.x & 31;
    int node = blockIdx.x * 8 + wave;
    if (node >= n) return;
    float sr = 0.0f, so = 0.0f;
#pragma unroll
    for (int j = 0; j < 4; ++j) {
        int c = lane + 32 * j;
        float y = stats[256 + c] * hpre[(size_t)node * HID_CH + c] + stats[384 + c];
        y = fmaxf(y, 0.0f);
        sr += y * w_rel2[c];
        so += y * w_root2[c];
    }
#pragma unroll
    for (int off = 16; off > 0; off >>= 1) {
        sr += __shfl_xor(sr, off, 32);
        so += __shfl_xor(so, off, 32);
    }
    if (lane == 0) {
        s_rel[node] = sr;
        out[node] = so + b_rel2[0];
    }
}

// ---------------------------------------------------------------------------
// 6) out[dst] += s_rel[src]  (1 scalar per edge — 128x less traffic than
//    aggregating in feature space; out fits in L2/WGP$)
// ---------------------------------------------------------------------------
__global__ void scatter_scalar_kernel(const float* __restrict__ s_rel, const int* __restrict__ ei,
                                      float* __restrict__ out, long long E) {
    long long e = (long long)blockIdx.x * blockDim.x + threadIdx.x;
    if (e >= E) return;
    atomicAdd(&out[ei[E + e]], s_rel[ei[e]]);
}

// ---------------------------------------------------------------------------
extern "C" void kernel_launch(void* const* d_in, const int* in_sizes, int n_in,
                              void* d_out, int out_size, void* d_ws, size_t ws_size,
                              hipStream_t stream) {
    const float* x       = (const float*)d_in[0];
    const int*   ei      = (const int*)  d_in[1];   // [2, E]: row0=src, row1=dst
    const float* w_rel1  = (const float*)d_in[2];
    const float* b_rel1  = (const float*)d_in[3];
    const float* w_root1 = (const float*)d_in[4];
    const float* gamma   = (const float*)d_in[5];
    const float* beta    = (const float*)d_in[6];
    const float* w_rel2  = (const float*)d_in[7];
    const float* b_rel2  = (const float*)d_in[8];
    const float* w_root2 = (const float*)d_in[9];

    const int n = in_sizes[0] / IN_CH;          // 100000
    const long long E = in_sizes[1] / 2;        // 1600000

    float* ws    = (float*)d_ws;
    float* agg   = ws;                               // n*64
    float* hpre  = agg + (size_t)n * IN_CH;          // n*128
    float* stats = hpre + (size_t)n * HID_CH;        // 512 (sum|sumsq|scale|shift)
    float* s_rel = stats + 512;                      // n
    float* out   = (float*)d_out;

    // 0) zero accumulators
    zero_ws_kernel<<<(n * IN_CH + 255) / 256, 256, 0, stream>>>(agg, stats, n * IN_CH);

    // 1) neighbor aggregation in 64-dim input space
    long long work1 = E * 64;
    scatter_feat_kernel<<<(unsigned)((work1 + 255) / 256), 256, 0, stream>>>(x, ei, agg, E);

    // 2) fused GraphConv-1 GEMM (WMMA f32)
    gemm1_wmma_kernel<<<(n + 127) / 128, 256, 0, stream>>>(agg, x, w_rel1, w_root1, b_rel1, hpre, n);

    // 3-4) batch-norm statistics + fold
    bn_stats_kernel<<<256, 256, 0, stream>>>(hpre, stats, n);
    bn_finalize_kernel<<<1, 128, 0, stream>>>(stats, gamma, beta, n);

    // 5) BN + ReLU + both layer-2 projections to per-node scalars
    bn_relu_proj_kernel<<<(n + 7) / 8, 256, 0, stream>>>(hpre, stats, w_rel2, w_root2, b_rel2,
                                                         s_rel, out, n);

    // 6) scalar edge aggregation
    scatter_scalar_kernel<<<(unsigned)((E + 255) / 256), 256, 0, stream>>>(s_rel, ei, out, E);
}